// ContrastiveLoss_23278722744784
// MI455X (gfx1250) — compile-verified
//
#include <hip/hip_runtime.h>
#include <math.h>

typedef __attribute__((ext_vector_type(2))) float v2f;
typedef __attribute__((ext_vector_type(8))) float v8f;

#define NB 64     // batch (images == captions)
#define NR 36     // regions
#define NW 50     // words (padded)
#define ND 1024   // feature dim
#define DK 128    // K-chunk staged through LDS
#define IST 132   // LDS row stride (132 mod 64 = 4 -> conflict-free strided b64 reads)
#define MARGIN 0.2f
#define LAM_SM 9.0f
#define LAM_LSE 6.0f
#define EPSF 1e-8f

__device__ __forceinline__ v2f ldv2g(const float* p) { return *(const v2f*)p; }
__device__ __forceinline__ v2f ldv2s(const float* p) { return *(const v2f*)p; }

// -------------------------------------------------------------------------
// Kernel 1: image Gram matrices G[b] = im_b @ im_b^T, padded to 48x48.
// One block per image b, 3 waves; wave = N-tile, each wave does 3 M-tiles.
// Pad rows are CLAMPED (not zeroed): the pad region of G is never read, so
// the inner loop carries no guards / exec-mask manipulation.
// -------------------------------------------------------------------------
__global__ void cl_gram_kernel(const float* __restrict__ im, float* __restrict__ gram) {
  const int b    = blockIdx.x;
  const int lane = threadIdx.x & 31;
  const int wave = threadIdx.x >> 5;      // 0..2 -> N tile
  const int half = lane >> 4;             // 0/1 (K sub-block)
  const int l15  = lane & 15;
  const float* imB = im + (size_t)b * NR * ND;

  const int n   = min(16 * wave + l15, NR - 1);  // clamped B-operand row
  const int m2  = min(32 + l15, NR - 1);         // clamped M-tile-2 row
  v8f a0 = {}, a1 = {}, a2 = {};

#pragma unroll 4
  for (int k = 0; k < ND; k += 4) {
    const int off = k + 2 * half;
    v2f bf = ldv2g(imB + (size_t)n         * ND + off);
    v2f f0 = ldv2g(imB + (size_t)(l15)     * ND + off);
    v2f f1 = ldv2g(imB + (size_t)(16 + l15)* ND + off);
    v2f f2 = ldv2g(imB + (size_t)m2        * ND + off);
    a0 = __builtin_amdgcn_wmma_f32_16x16x4_f32(false, f0, false, bf, (short)0, a0, false, false);
    a1 = __builtin_amdgcn_wmma_f32_16x16x4_f32(false, f1, false, bf, (short)0, a1, false, false);
    a2 = __builtin_amdgcn_wmma_f32_16x16x4_f32(false, f2, false, bf, (short)0, a2, false, false);
  }

  float* Gb = gram + (size_t)b * 48 * 48;
  const int col = 16 * wave + l15;
#pragma unroll
  for (int v = 0; v < 8; ++v) {
    const int row = v + 8 * half;
    Gb[(row     ) * 48 + col] = a0[v];
    Gb[(row + 16) * 48 + col] = a1[v];
    Gb[(row + 32) * 48 + col] = a2[v];
  }
}

// -------------------------------------------------------------------------
// Kernel 2: caption word L2 norms cn[c][w] = ||s[c,w,:]||.
// -------------------------------------------------------------------------
__global__ void cl_capnorm_kernel(const float* __restrict__ s, float* __restrict__ cn) {
  const int c = blockIdx.x;
  const int w = threadIdx.x;
  if (w >= NW) return;
  const float* row = s + ((size_t)c * NW + w) * ND;
  float ss = 0.f;
  for (int d = 0; d < ND; d += 4) {
    float4 v = *(const float4*)(row + d);
    ss += v.x * v.x + v.y * v.y + v.z * v.z + v.w * v.w;
  }
  cn[c * NW + w] = sqrtf(ss);
}

// -------------------------------------------------------------------------
// Kernel 3: per-(caption c, image b) score.
// Phase A: A = im_b (48x1024, clamped pad) @ s_c^T (1024x64, clamped pad)
//          as a WMMA f32 16x16x4 GEMM. K is streamed in DK=128 chunks
//          staged through LDS with coalesced b128 loads; fragments are
//          ds_load_b64 with a 132-float row stride (all 64 banks hit).
// Phase B: clipped-L2 row norm, softmax over regions, num, Sm^T G Sm,
//          cosine + masked LSE -> scores[b][c].
// LDS for the two phases is disjoint in time -> union.
// -------------------------------------------------------------------------
__global__ void __launch_bounds__(128)
cl_score_kernel(const float* __restrict__ im,
                const float* __restrict__ s,
                const int*   __restrict__ s_l,
                const float* __restrict__ gram,
                const float* __restrict__ cn,
                float* __restrict__ scores) {
  __shared__ union {
    struct {                      // GEMM phase
      float imS[NR * IST];        // 36 rows x 132
      float sS[NW * IST];         // 50 rows x 132
    } g;
    struct {                      // post phase
      float As[48 * 65];
      float An[36 * 65];
      float Gs[36 * 37];
      float Smb[50 * 37];
      float red[64];
    } p;
  } sh;

  const int c = blockIdx.x;       // caption index
  const int b = blockIdx.y;       // image index
  const int tid  = threadIdx.x;
  const int lane = tid & 31;
  const int wave = tid >> 5;      // 0..3 -> word (N) tile
  const int half = lane >> 4;
  const int l15  = lane & 15;

  const float* imB = im + (size_t)b * NR * ND;
  const float* sC  = s  + (size_t)c * NW * ND;

  const int mr0 = l15;                       // < 36
  const int mr1 = 16 + l15;                  // < 36
  const int mr2 = min(32 + l15, NR - 1);     // clamped pad row
  const int wr  = min(16 * wave + l15, NW - 1); // clamped pad word
  v8f a0 = {}, a1 = {}, a2 = {};

  for (int k0 = 0; k0 < ND; k0 += DK) {
    // ---- stage K-chunk of im_b and s_c into LDS (coalesced b128) ----
    for (int i = tid; i < NR * (DK / 4); i += 128) {
      const int row = i / (DK / 4), q = i % (DK / 4);
      float4 v = *(const float4*)(imB + (size_t)row * ND + k0 + q * 4);
      *(float4*)(&sh.g.imS[row * IST + q * 4]) = v;
    }
    for (int i = tid; i < NW * (DK / 4); i += 128) {
      const int row = i / (DK / 4), q = i % (DK / 4);
      float4 v = *(const float4*)(sC + (size_t)row * ND + k0 + q * 4);
      *(float4*)(&sh.g.sS[row * IST + q * 4]) = v;
    }
    // hint the next chunk toward L2/L0 while we compute this one
    if (k0 + DK < ND) {
      __builtin_prefetch(imB + (size_t)(tid % NR) * ND + k0 + DK + (tid / NR) * 32, 0, 0);
      __builtin_prefetch(sC  + (size_t)(tid % NW) * ND + k0 + DK + (tid / NW) * 32, 0, 0);
    }
    __syncthreads();

    // ---- WMMA over the chunk: 3 M-tiles per wave, fragments from LDS ----
#pragma unroll 4
    for (int kk = 0; kk < DK; kk += 4) {
      const int off = kk + 2 * half;
      v2f bf = ldv2s(&sh.g.sS [wr  * IST + off]);
      v2f f0 = ldv2s(&sh.g.imS[mr0 * IST + off]);
      v2f f1 = ldv2s(&sh.g.imS[mr1 * IST + off]);
      v2f f2 = ldv2s(&sh.g.imS[mr2 * IST + off]);
      a0 = __builtin_amdgcn_wmma_f32_16x16x4_f32(false, f0, false, bf, (short)0, a0, false, false);
      a1 = __builtin_amdgcn_wmma_f32_16x16x4_f32(false, f1, false, bf, (short)0, a1, false, false);
      a2 = __builtin_amdgcn_wmma_f32_16x16x4_f32(false, f2, false, bf, (short)0, a2, false, false);
    }
    __syncthreads();   // chunk fully consumed before restage / phase B
  }

  // ---- spill C tiles to LDS (stride 65 vs banks) ----
  {
    const int col = 16 * wave + l15;
#pragma unroll
    for (int v = 0; v < 8; ++v) {
      const int row = v + 8 * half;
      sh.p.As[(row     ) * 65 + col] = a0[v];
      sh.p.As[(row + 16) * 65 + col] = a1[v];
      sh.p.As[(row + 32) * 65 + col] = a2[v];
    }
  }
  // stage Gram for this image (only the real 36x36 region)
  {
    const float* Gb = gram + (size_t)b * 48 * 48;
    for (int i = tid; i < NR * NR; i += blockDim.x) {
      const int r = i / NR, rr = i % NR;
      sh.p.Gs[r * 37 + rr] = Gb[r * 48 + rr];
    }
  }
  __syncthreads();

  const int L = s_l[c];

  // ---- clipped_l2norm: LeakyReLU(0.1), mask, L2-normalize over words ----
  if (tid < NR) {
    float ss = 0.f;
    for (int ww = 0; ww < L; ++ww) {
      float a  = sh.p.As[tid * 65 + ww];
      float al = (a > 0.f) ? a : 0.1f * a;
      ss += al * al;
    }
    const float inv = 1.0f / (sqrtf(ss) + EPSF);
    for (int ww = 0; ww < NW; ++ww) {
      float a  = sh.p.As[tid * 65 + ww];
      float al = (a > 0.f) ? a : 0.1f * a;
      sh.p.An[tid * 65 + ww] = (ww < L) ? al * inv : 0.f;
    }
  }
  __syncthreads();

  // ---- per-word: softmax over regions, num, Sm^T G Sm, cosine, exp ----
  float e = 0.f;
  if (tid < NW && tid < L) {
    const int ww = tid;
    float mx = -1e30f;
    for (int r = 0; r < NR; ++r) mx = fmaxf(mx, sh.p.An[r * 65 + ww] * LAM_SM);
    float ssum = 0.f;
    for (int r = 0; r < NR; ++r) {
      float ev = __expf(sh.p.An[r * 65 + ww] * LAM_SM - mx);
      sh.p.Smb[ww * 37 + r] = ev;
      ssum += ev;
    }
    const float invs = 1.f / ssum;
    float num = 0.f;
    for (int r = 0; r < NR; ++r) {
      float smr = sh.p.Smb[ww * 37 + r] * invs;
      sh.p.Smb[ww * 37 + r] = smr;
      num += smr * sh.p.As[r * 65 + ww];     // sum_r Sm * A (raw A)
    }
    float wsq = 0.f;
    for (int r = 0; r < NR; ++r) {
      float t = 0.f;
      for (int rr = 0; rr < NR; ++rr) t += sh.p.Gs[r * 37 + rr] * sh.p.Smb[ww * 37 + rr];
      wsq += sh.p.Smb[ww * 37 + r] * t;      // Sm^T G Sm
    }
    const float wn  = sqrtf(fmaxf(wsq, 0.f));
    const float cnw = cn[c * NW + ww];
    const float sim = num / fmaxf(cnw * wn, EPSF);
    e = __expf(sim * LAM_LSE);
  }
  if (tid < 64) sh.p.red[tid] = e;
  __syncthreads();

  if (tid == 0) {
    float acc = 0.f;
    for (int i = 0; i < 64; ++i) acc += sh.p.red[i];
    scores[b * NB + c] = __logf(acc) / LAM_LSE;   // scores[image][caption]
  }
}

// -------------------------------------------------------------------------
// Kernel 4: hardest-negative hinge loss over the 64x64 score matrix.
// -------------------------------------------------------------------------
__global__ void cl_loss_kernel(const float* __restrict__ scores, float* __restrict__ out) {
  __shared__ float S[64 * 65];
  __shared__ float partial[64];
  const int t = threadIdx.x;
  for (int i = t; i < 64 * 64; i += 64) S[(i >> 6) * 65 + (i & 63)] = scores[i];
  __syncthreads();

  const float di = S[t * 65 + t];
  float rmax = 0.f, cmax = 0.f;     // relu + zeroed diagonal folded into init
  for (int j = 0; j < 64; ++j) {
    if (j != t) {
      float cs = MARGIN + S[t * 65 + j] - di;  // cost_s row t
      if (cs > rmax) rmax = cs;
      float ci = MARGIN + S[j * 65 + t] - di;  // cost_im col t
      if (ci > cmax) cmax = ci;
    }
  }
  partial[t] = rmax + cmax;
  __syncthreads();
  if (t == 0) {
    float acc = 0.f;
    for (int i = 0; i < 64; ++i) acc += partial[i];
    out[0] = acc;
  }
}

// -------------------------------------------------------------------------
extern "C" void kernel_launch(void* const* d_in, const int* in_sizes, int n_in,
                              void* d_out, int out_size, void* d_ws, size_t ws_size,
                              hipStream_t stream) {
  const float* im  = (const float*)d_in[0];   // (64, 36, 1024) f32
  const float* s   = (const float*)d_in[1];   // (64, 50, 1024) f32
  const int*   s_l = (const int*)d_in[2];     // (64,) i32

  float* gram   = (float*)d_ws;               // 64 * 48 * 48
  float* cn     = gram + (size_t)NB * 48 * 48;// 64 * 50
  float* scores = cn + (size_t)NB * NW;       // 64 * 64

  cl_gram_kernel   <<<dim3(NB),        96, 0, stream>>>(im, gram);
  cl_capnorm_kernel<<<dim3(NB),        64, 0, stream>>>(s, cn);
  cl_score_kernel  <<<dim3(NB, NB),   128, 0, stream>>>(im, s, s_l, gram, cn, scores);
  cl_loss_kernel   <<<1,               64, 0, stream>>>(scores, (float*)d_out);
}